// ERoPE_47012712022630
// MI455X (gfx1250) — compile-verified
//
#include <hip/hip_runtime.h>

// Problem constants (match reference): B=2, N=512, M=50, P=2, L=9, D=32
#define B_ 2
#define N_ 512
#define M_ 50
#define T_ 576           // P*L*D — flattened q/k length == flattened v length
#define NT16 (N_ / 16)   // 32 row-tiles per batch

typedef float v2f __attribute__((ext_vector_type(2)));
typedef float v4f __attribute__((ext_vector_type(4)));
typedef float v8f __attribute__((ext_vector_type(8)));

// ---------------------------------------------------------------------------
// Phase 1: dot[b,n] = sum_t q[b,n,t]*k[b,n,t] via V_WMMA_F32_16X16X4_F32.
// One wave (32 lanes, EXEC all-1s) per 16-row tile. A = q-tile (16xK),
// B = k-tile transposed (Kx16); diagonal of the 16x16 accumulator holds the
// per-row dot products. 144 WMMA k-steps of 4.
//
// 16x16x4 f32 operand layout (ISA 7.12.2):
//   A (16x4):  lanes 0-15 -> M=lane, VGPR0=K0,VGPR1=K1; lanes 16-31 -> K2,K3
//   B (4x16):  lanes 0-15 -> N=lane, VGPR0=K0,VGPR1=K1; lanes 16-31 -> K2,K3
// so both operands load as row[kb + (lane<16?0:2) + {0,1}] — one b64 load each.
//   C/D (16x16): VGPR r: lanes 0-15 -> D[r][lane]; lanes 16-31 -> D[8+r][lane-16]
// ---------------------------------------------------------------------------
__global__ __launch_bounds__(32) void qk_dot_wmma(const float* __restrict__ q,
                                                  const float* __restrict__ k,
                                                  float* __restrict__ dotv) {
    const int tile = blockIdx.x;              // [0, B_*NT16)
    const int b    = tile >> 5;               // NT16 == 32
    const int nb   = (tile & 31) << 4;        // first row of this 16-row tile
    const int lane = threadIdx.x;             // wave32
    const int row  = nb + (lane & 15);
    const int koff = (lane < 16) ? 0 : 2;

    const float* __restrict__ qrow = q + (size_t)(b * N_ + row) * T_ + koff;
    const float* __restrict__ krow = k + (size_t)(b * N_ + row) * T_ + koff;

    v8f acc = {};
#pragma unroll 4
    for (int kb = 0; kb < T_; kb += 4) {
        v2f a  = *(const v2f*)(qrow + kb);
        v2f bm = *(const v2f*)(krow + kb);
        // 8 args: (neg_a, A, neg_b, B, c_mod, C, reuse_a, reuse_b)
        acc = __builtin_amdgcn_wmma_f32_16x16x4_f32(false, a, false, bm,
                                                    (short)0, acc, false, false);
    }

    // Harvest the diagonal: D[i][i], i<8 -> (vgpr i, lane i);
    //                       D[8+r][8+r] -> (vgpr r, lane 24+r).
    float* dst = dotv + b * N_ + nb;
#pragma unroll
    for (int r = 0; r < 8; ++r) {
        if (lane == r)      dst[r]     = acc[r];
        if (lane == r + 24) dst[r + 8] = acc[r];
    }
}

// ---------------------------------------------------------------------------
// Phase 2 (bandwidth-bound, ~118 MB of stores):
//   y[b,m,n,:] = (dot[b,n]/24) * w[b,m] * v[b,n,:]
// One 128-thread block per output row (144 float4 chunks). v-row comes from
// L2/WGP$ (2.4 MB resident, reused 50x); y goes out with non-temporal B128
// stores so the stream doesn't wash caches.
// ---------------------------------------------------------------------------
__global__ __launch_bounds__(128) void bcast_scale(const float* __restrict__ v,
                                                   const float* __restrict__ dotv,
                                                   const float* __restrict__ gw,
                                                   float* __restrict__ y) {
    const int row = blockIdx.x;               // (b*M + m)*N + n
    const int n   = row & (N_ - 1);           // N_ = 512
    const int bm  = row >> 9;                 // b*M + m
    const int b   = bm / M_;

    const float s = dotv[b * N_ + n] * gw[bm] * (1.0f / 24.0f);  // 1/sqrt(576)

    const v4f* __restrict__ vr = (const v4f*)(v + (size_t)(b * N_ + n) * T_);
    v4f* __restrict__ yr       = (v4f*)(y + (size_t)row * T_);

    const int t = threadIdx.x;

    v4f a  = vr[t];
    __builtin_nontemporal_store(a * s, yr + t);
    if (t < (T_ / 4 - 128)) {                 // chunks 128..143
        v4f a2 = vr[t + 128];
        __builtin_nontemporal_store(a2 * s, yr + t + 128);
    }
}

extern "C" void kernel_launch(void* const* d_in, const int* in_sizes, int n_in,
                              void* d_out, int out_size, void* d_ws, size_t ws_size,
                              hipStream_t stream) {
    // setup_inputs order: q, k, v, pos, grid_u, grid_w, theta
    const float* q  = (const float*)d_in[0];
    const float* k  = (const float*)d_in[1];
    const float* v  = (const float*)d_in[2];
    const float* gw = (const float*)d_in[5];
    // pos/grid_u/theta are mathematically irrelevant: the eRoPE rotation is
    // orthogonal and applied identically to q and k, so qr·kr == q·k.

    float* y    = (float*)d_out;
    float* dotv = (float*)d_ws;               // B_*N_ floats = 4 KB scratch

    qk_dot_wmma<<<B_ * NT16, 32, 0, stream>>>(q, k, dotv);
    bcast_scale<<<B_ * M_ * N_, 128, 0, stream>>>(v, dotv, gw, y);
}